// NodeAttention_12223476925026
// MI455X (gfx1250) — compile-verified
//
#include <hip/hip_runtime.h>
#include <math.h>

// B=16, N=2048, D_IN=D_OUT=128
#define B_   16
#define N_   2048
#define D_   128
#define JT   32      // j-chunk staged in LDS for the attention GEMM
#define PSTR 288     // dwords per K-pair row in LDS (128*2 + 32 pad -> disjoint bank halves)

typedef float v2f __attribute__((ext_vector_type(2)));
typedef float v8f __attribute__((ext_vector_type(8)));

static __device__ __forceinline__ v8f wmma_f32_4(v2f a, v2f b, v8f c) {
  // D = A(16x4 f32) * B(4x16 f32) + C(16x16 f32)
  return __builtin_amdgcn_wmma_f32_16x16x4_f32(false, a, false, b, (short)0, c,
                                               false, false);
}

// ---------------------------------------------------------------- kernel 0
// K-pair interleaved transpose:  Wt2[((k>>1)*D + d)*2 + (k&1)] = W[d][k]
// -> a lane's (K, K+1) B-operand pair is one contiguous b64.
__global__ __launch_bounds__(256) void k_transpose_w(const float* __restrict__ W,
                                                     float* __restrict__ Wt2) {
  int idx = blockIdx.x * 256 + threadIdx.x;   // 0 .. 128*128-1
  int d = idx >> 7, k = idx & 127;
  Wt2[((k >> 1) * D_ + d) * 2 + (k & 1)] = W[idx];
}

// ---------------------------------------------------------------- kernel 1
// seq[b,i,d] = sum_k x[b,i,k] * W[d,k] + bias[d]
// 1 wave -> 16 rows x 128 cols; A as b64 from x, B as b64 from pair-interleaved Wt2.
__global__ __launch_bounds__(256) void k_emb(const float* __restrict__ x,
                                             const float* __restrict__ Wt2,
                                             const float* __restrict__ bias,
                                             float* __restrict__ seq) {
  int b    = blockIdx.y;
  int i0   = blockIdx.x * 128;
  int tid  = threadIdx.x;
  int wave = tid >> 5, lane = tid & 31;
  int m    = lane & 15;                 // A-row / B-col within tile
  int kb   = (lane & 16) ? 2 : 0;       // K base for this lane half
  int row  = i0 + wave * 16 + m;
  const float* xrow = x + ((size_t)b * N_ + row) * D_;

  v8f acc[8];
#pragma unroll
  for (int t = 0; t < 8; ++t)
#pragma unroll
    for (int e = 0; e < 8; ++e) acc[t][e] = 0.f;

#pragma unroll 4
  for (int kk = 0; kk < D_; kk += 4) {
    v2f a = *(const v2f*)(xrow + kk + kb);                       // b64
    const v2f* wp = (const v2f*)(Wt2 + (size_t)((kk + kb) >> 1) * (2 * D_));
#pragma unroll
    for (int t = 0; t < 8; ++t)
      acc[t] = wmma_f32_4(a, wp[t * 16 + m], acc[t]);            // b64 B feed
  }

  int n    = lane & 15;
  int half = (lane & 16) ? 8 : 0;
#pragma unroll
  for (int t = 0; t < 8; ++t) {
    int d    = t * 16 + n;
    float bi = bias[d];
#pragma unroll
    for (int r = 0; r < 8; ++r) {
      int orow = i0 + wave * 16 + half + r;
      seq[((size_t)b * N_ + orow) * D_ + d] = acc[t][r] + bi;
    }
  }
}

// ---------------------------------------------------------------- kernel 2
// q[b,n] = seq[b,n,:] . w_q + b_q ;  k likewise
__global__ __launch_bounds__(256) void k_qk(const float* __restrict__ seq,
                                            const float* __restrict__ wq,
                                            const float* __restrict__ bq,
                                            const float* __restrict__ wk,
                                            const float* __restrict__ bk,
                                            float* __restrict__ qv,
                                            float* __restrict__ kv) {
  int gid = blockIdx.x * 256 + threadIdx.x;   // 0 .. B*N-1
  const float* r = seq + (size_t)gid * D_;
  float sq = 0.f, sk = 0.f;
#pragma unroll 8
  for (int d = 0; d < D_; ++d) {
    float v = r[d];
    sq = fmaf(v, wq[d], sq);
    sk = fmaf(v, wk[d], sk);
  }
  qv[gid] = sq + bq[0];
  kv[gid] = sk + bk[0];
}

// ---------------------------------------------------------------- kernel 3
// Per-column softmax stats over i:  g[i,j] = leaky(adj[i,j]*(k[i]+q[j]))
// mcol[b,j] = max_i g ; rscol[b,j] = 1 / sum_i exp(g - m)
__global__ __launch_bounds__(256) void k_stats(const float* __restrict__ adj,
                                               const float* __restrict__ qv,
                                               const float* __restrict__ kv,
                                               float* __restrict__ mcol,
                                               float* __restrict__ rscol) {
  int b = blockIdx.y;
  int j = blockIdx.x * 256 + threadIdx.x;
  float qj = qv[b * N_ + j];
  const float* kb = kv + b * N_;
  float mm = -INFINITY, ss = 0.f;
  for (int i = 0; i < N_; ++i) {
    float a = adj[(size_t)i * N_ + j];          // coalesced over j
    float t = a * (kb[i] + qj);
    float g = t > 0.f ? t : 0.01f * t;
    float nm = fmaxf(mm, g);
    ss = ss * __expf(mm - nm) + __expf(g - nm); // online softmax
    mm = nm;
  }
  mcol[b * N_ + j]  = mm;
  rscol[b * N_ + j] = 1.f / ss;
}

// ---------------------------------------------------------------- kernel 4
// out[b,i,d] = sum_j exp(g[i,j]-m[j])*rs[j] * seq[b,j,d]
// Block: 8 waves x 32 rows = 256 rows. LDS: double-buffered 32x128 seq chunk,
// K-pair interleaved (b64 reads, conflict-free). 2 A-tiles per wave share B.
__global__ __launch_bounds__(256) void k_attn(const float* __restrict__ adj,
                                              const float* __restrict__ seq,
                                              const float* __restrict__ qv,
                                              const float* __restrict__ kv,
                                              const float* __restrict__ mcol,
                                              const float* __restrict__ rscol,
                                              float* __restrict__ out) {
  __shared__ __align__(16) float Bs[2][16 * PSTR];   // 2 x 18.4 KB
  int b    = blockIdx.y;
  int i0   = blockIdx.x * 256;
  int tid  = threadIdx.x;
  int wave = tid >> 5, lane = tid & 31;
  int m    = lane & 15;
  int kb   = (lane & 16) ? 2 : 0;
  int row0 = i0 + wave * 32 + m;
  int row1 = row0 + 16;
  float krow0 = kv[b * N_ + row0];
  float krow1 = kv[b * N_ + row1];
  const float* adjr0 = adj + (size_t)row0 * N_;
  const float* adjr1 = adj + (size_t)row1 * N_;
  const float* qb = qv    + b * N_;
  const float* mb = mcol  + b * N_;
  const float* rb = rscol + b * N_;
  const float* seqb = seq + (size_t)b * N_ * D_;

  // staging task: idx = u*256+tid -> pair p = idx>>5 (u<2: p 0..15 over both u),
  // quad q4 = idx&31 ... use: 16 pairs * 32 d-quads = 512 tasks, 2 per thread.
  int sp[2], sd[2];
#pragma unroll
  for (int u = 0; u < 2; ++u) {
    int idx = u * 256 + tid;
    sp[u] = idx >> 5;          // pair 0..15
    sd[u] = (idx & 31) * 4;    // d 0..124 (quad)
  }

  float4 ra[2], rc[2];
  // load chunk j0 rows (2p, 2p+1), 4 d's each, into registers
  auto ldnext = [&](int j0) {
#pragma unroll
    for (int u = 0; u < 2; ++u) {
      const float* r0 = seqb + (size_t)(j0 + 2 * sp[u]) * D_ + sd[u];
      ra[u] = *(const float4*)r0;
      rc[u] = *(const float4*)(r0 + D_);
    }
  };
  // interleave (K,K+1) and store as two b128s
  auto stnext = [&](int buf) {
#pragma unroll
    for (int u = 0; u < 2; ++u) {
      float* p = &Bs[buf][sp[u] * PSTR + 2 * sd[u]];
      float4 s0 = make_float4(ra[u].x, rc[u].x, ra[u].y, rc[u].y);
      float4 s1 = make_float4(ra[u].z, rc[u].z, ra[u].w, rc[u].w);
      ((float4*)p)[0] = s0;
      ((float4*)p)[1] = s1;
    }
  };

  v8f acc[2][8];
#pragma unroll
  for (int s = 0; s < 2; ++s)
#pragma unroll
    for (int t = 0; t < 8; ++t)
#pragma unroll
      for (int e = 0; e < 8; ++e) acc[s][t][e] = 0.f;

  ldnext(0);
  stnext(0);

  for (int j0 = 0; j0 < N_; j0 += JT) {
    int cur = (j0 >> 5) & 1;
    bool have_next = (j0 + JT) < N_;
    if (have_next) ldnext(j0 + JT);     // issue global loads early (overlap)
    __syncthreads();                    // Bs[cur] ready; prev reads done

#pragma unroll
    for (int kk = 0; kk < JT; kk += 4) {
      int jA = j0 + kk + kb;            // uniform per lane half
      // per-j scalars shared by both A-tiles
      float q0 = qb[jA], q1 = qb[jA + 1];
      float m0 = mb[jA], m1 = mb[jA + 1];
      float r0 = rb[jA], r1 = rb[jA + 1];
      float a00 = adjr0[jA], a01 = adjr0[jA + 1];
      float a10 = adjr1[jA], a11 = adjr1[jA + 1];
      float t00 = a00 * (krow0 + q0), t01 = a01 * (krow0 + q1);
      float t10 = a10 * (krow1 + q0), t11 = a11 * (krow1 + q1);
      float g00 = t00 > 0.f ? t00 : 0.01f * t00;
      float g01 = t01 > 0.f ? t01 : 0.01f * t01;
      float g10 = t10 > 0.f ? t10 : 0.01f * t10;
      float g11 = t11 > 0.f ? t11 : 0.01f * t11;
      v2f aA, aB;
      aA.x = __expf(g00 - m0) * r0;  aA.y = __expf(g01 - m1) * r1;
      aB.x = __expf(g10 - m0) * r0;  aB.y = __expf(g11 - m1) * r1;

      const v2f* bp = (const v2f*)(&Bs[cur][(((kk + kb) >> 1)) * PSTR]);
#pragma unroll
      for (int t = 0; t < 8; ++t) {
        v2f bv = bp[t * 16 + m];        // ds_load_b64, conflict-free
        acc[0][t] = wmma_f32_4(aA, bv, acc[0][t]);
        acc[1][t] = wmma_f32_4(aB, bv, acc[1][t]);
      }
    }

    if (have_next) stnext(cur ^ 1);     // fill other buffer after compute
  }

  int n    = lane & 15;
  int half = (lane & 16) ? 8 : 0;
#pragma unroll
  for (int s = 0; s < 2; ++s)
#pragma unroll
    for (int t = 0; t < 8; ++t)
#pragma unroll
      for (int r = 0; r < 8; ++r) {
        int orow = i0 + wave * 32 + s * 16 + half + r;
        out[((size_t)b * N_ + orow) * D_ + t * 16 + n] = acc[s][t][r];
      }
}

// ---------------------------------------------------------------- launcher
extern "C" void kernel_launch(void* const* d_in, const int* in_sizes, int n_in,
                              void* d_out, int out_size, void* d_ws, size_t ws_size,
                              hipStream_t stream) {
  const float* x    = (const float*)d_in[0];
  const float* adj  = (const float*)d_in[1];
  const float* Wemb = (const float*)d_in[2];
  const float* bemb = (const float*)d_in[3];
  const float* wq   = (const float*)d_in[4];
  const float* bq   = (const float*)d_in[5];
  const float* wk   = (const float*)d_in[6];
  const float* bk   = (const float*)d_in[7];
  float* outp = (float*)d_out;

  float* ws   = (float*)d_ws;
  float* seq  = ws;                               // B*N*D = 4,194,304 floats
  float* qv   = seq  + (size_t)B_ * N_ * D_;      // 32768
  float* kv   = qv   + (size_t)B_ * N_;           // 32768
  float* mcol = kv   + (size_t)B_ * N_;           // 32768
  float* rsc  = mcol + (size_t)B_ * N_;           // 32768
  float* Wt2  = rsc  + (size_t)B_ * N_;           // 16384 (pair-interleaved W^T)

  k_transpose_w<<<(D_ * D_) / 256, 256, 0, stream>>>(Wemb, Wt2);
  k_emb<<<dim3(N_ / 128, B_), 256, 0, stream>>>(x, Wt2, bemb, seq);
  k_qk<<<(B_ * N_) / 256, 256, 0, stream>>>(seq, wq, bq, wk, bk, qv, kv);
  k_stats<<<dim3(N_ / 256, B_), 256, 0, stream>>>(adj, qv, kv, mcol, rsc);
  k_attn<<<dim3(N_ / 256, B_), 256, 0, stream>>>(adj, seq, qv, kv, mcol, rsc, outp);
}